// ResidualPointNet_63307817943141
// MI455X (gfx1250) — compile-verified
//
#include <hip/hip_runtime.h>

// ---------------------------------------------------------------------------
// ResidualPointNet forward for MI455X (gfx1250), fp32 end-to-end.
// All GEMMs use V_WMMA_F32_16X16X4_F32 (wave32). Each wave computes a 64x16
// output strip (4 accumulators) so one B fetch feeds 4 WMMAs.
// ---------------------------------------------------------------------------

#define BB 8      // batch
#define NN 2048   // points
#define CC 256    // channel width C
#define HH 128    // hidden H
#define CQ 64     // q/k dim C/4

typedef __attribute__((ext_vector_type(2))) float v2f;
typedef __attribute__((ext_vector_type(8))) float v8f;

enum { EPI_NONE = 0, EPI_BIAS = 1, EPI_BIAS_RELU = 2, EPI_BIAS_ADD_RELU = 3, EPI_SCALE_COL = 4 };

// ---------------------------------------------------------------------------
// Generic fp32 WMMA GEMM:  Y[M,NN] = A[M,K] * B[K,NN]  (+ fused epilogue)
//   A row-major (lda) or "transposed" meaning A(m,k) = A[k*lda + m]
//   B row-major with row stride NN (columns contiguous).
//   grid = (NN/64, M/64, nbatch), block = 128 (4 waves).
//   Wave w covers columns (blockIdx.x*4+w)*16..+16 and rows blockIdx.y*64..+64.
// ---------------------------------------------------------------------------
__global__ __launch_bounds__(128) void wgemm_f32(
    const float* __restrict__ A, const float* __restrict__ Bm, float* __restrict__ Y,
    const float* __restrict__ bias, const float* __restrict__ addt,
    const float* __restrict__ scale,
    int K, int lda, int a_trans,
    long a_bs, long b_bs, long y_bs, long add_bs, int epi)
{
  const int wid  = threadIdx.x >> 5;
  const int lane = threadIdx.x & 31;
  const int l15  = lane & 15;
  const int hi   = lane >> 4;
  const int n0   = (blockIdx.x * 4 + wid) * 16;
  const int m0   = blockIdx.y * 64;
  const int col  = n0 + l15;
  const long ab = (long)blockIdx.z * a_bs;
  const long bb = (long)blockIdx.z * b_bs;
  const long yb = (long)blockIdx.z * y_bs;

  v8f acc[4];
#pragma unroll
  for (int t = 0; t < 4; ++t) acc[t] = (v8f){0.f, 0.f, 0.f, 0.f, 0.f, 0.f, 0.f, 0.f};

  const float* Bp = Bm + bb + col;

  if (!a_trans) {
    const float* Ap = A + ab + (long)(m0 + l15) * lda;
    for (int k0 = 0; k0 < K; k0 += 4) {
      const int kk = k0 + 2 * hi;
      v2f b;
      b.x = Bp[(long)kk * NN];                   // B(kk, col)
      b.y = Bp[(long)(kk + 1) * NN];             // B(kk+1, col)
#pragma unroll
      for (int t = 0; t < 4; ++t) {
        v2f a = *(const v2f*)(Ap + (long)t * 16 * lda + kk);   // A(m0+t*16+l15, kk..kk+1)
        acc[t] = __builtin_amdgcn_wmma_f32_16x16x4_f32(false, a, false, b,
                                                       (short)0, acc[t], false, false);
      }
    }
  } else {
    const float* Ap = A + ab + (m0 + l15);       // A(m,k) = A[k*lda + m]
    for (int k0 = 0; k0 < K; k0 += 4) {
      const int kk = k0 + 2 * hi;
      v2f b;
      b.x = Bp[(long)kk * NN];
      b.y = Bp[(long)(kk + 1) * NN];
      const float* Ar0 = Ap + (long)kk * lda;
      const float* Ar1 = Ap + (long)(kk + 1) * lda;
#pragma unroll
      for (int t = 0; t < 4; ++t) {
        v2f a;
        a.x = Ar0[t * 16];
        a.y = Ar1[t * 16];
        acc[t] = __builtin_amdgcn_wmma_f32_16x16x4_f32(false, a, false, b,
                                                       (short)0, acc[t], false, false);
      }
    }
  }

  const float sc = (epi == EPI_SCALE_COL) ? scale[col] : 1.0f;
#pragma unroll
  for (int t = 0; t < 4; ++t) {
#pragma unroll
    for (int g = 0; g < 8; ++g) {
      const int row = m0 + t * 16 + g + hi * 8;  // C/D layout: VGPR g -> rows g, g+8
      float v = acc[t][g];
      if (epi == EPI_BIAS || epi == EPI_BIAS_RELU || epi == EPI_BIAS_ADD_RELU) v += bias[row];
      if (epi == EPI_BIAS_ADD_RELU) v += addt[(long)blockIdx.z * add_bs + (long)row * NN + col];
      if (epi == EPI_BIAS_RELU || epi == EPI_BIAS_ADD_RELU) v = fmaxf(v, 0.f);
      Y[yb + (long)row * NN + col] = v * sc;
    }
  }
}

// ---------------------------------------------------------------------------
// 3-tap conv (GCA "w3"): Y[b,o,n] = sum_i sum_t w3[o,i,t] * low[b,i,n+t-1]
// K = CC*3 = 768 GEMM with shifted-column B loader. grid=(NN/64, CC/64, BB)
// ---------------------------------------------------------------------------
__global__ __launch_bounds__(128) void wgemm_conv3(
    const float* __restrict__ W3, const float* __restrict__ X, float* __restrict__ Y)
{
  const int wid  = threadIdx.x >> 5;
  const int lane = threadIdx.x & 31;
  const int l15  = lane & 15;
  const int hi   = lane >> 4;
  const int n0   = (blockIdx.x * 4 + wid) * 16;
  const int m0   = blockIdx.y * 64;
  const int col  = n0 + l15;
  const long xb  = (long)blockIdx.z * CC * NN;
  const long yb  = (long)blockIdx.z * CC * NN;

  v8f acc[4];
#pragma unroll
  for (int t = 0; t < 4; ++t) acc[t] = (v8f){0.f, 0.f, 0.f, 0.f, 0.f, 0.f, 0.f, 0.f};

  const float* Ap = W3 + (long)(m0 + l15) * 768;

  for (int k0 = 0; k0 < 768; k0 += 4) {
    const int kk = k0 + 2 * hi;
    const int ka = kk, kb = kk + 1;
    const int i0 = ka / 3, t0 = ka - 3 * i0;
    const int i1 = kb / 3, t1 = kb - 3 * i1;
    const int c0 = col + t0 - 1;
    const int c1 = col + t1 - 1;
    v2f b;
    b.x = (c0 >= 0 && c0 < NN) ? X[xb + (long)i0 * NN + c0] : 0.f;
    b.y = (c1 >= 0 && c1 < NN) ? X[xb + (long)i1 * NN + c1] : 0.f;
#pragma unroll
    for (int t = 0; t < 4; ++t) {
      v2f a = *(const v2f*)(Ap + (long)t * 16 * 768 + kk);
      acc[t] = __builtin_amdgcn_wmma_f32_16x16x4_f32(false, a, false, b,
                                                     (short)0, acc[t], false, false);
    }
  }
#pragma unroll
  for (int t = 0; t < 4; ++t)
#pragma unroll
    for (int g = 0; g < 8; ++g)
      Y[yb + (long)(m0 + t * 16 + g + hi * 8) * NN + col] = acc[t][g];
}

// ---------------------------------------------------------------------------
// Small helper kernels
// ---------------------------------------------------------------------------
__global__ void fc_in_k(const float* __restrict__ x, const float* __restrict__ pos,
                        const float* __restrict__ w, const float* __restrict__ bias,
                        float* __restrict__ out)
{
  const long i = (long)blockIdx.x * 256 + threadIdx.x;       // over B*CC*NN
  const int n = (int)(i % NN);
  const int o = (int)((i / NN) % CC);
  const int b = (int)(i / ((long)NN * CC));
  const float* wr = w + o * 6;
  float s = bias[o];
#pragma unroll
  for (int c = 0; c < 3; ++c) s += wr[c]     * x  [((long)b * 3 + c) * NN + n];
#pragma unroll
  for (int c = 0; c < 3; ++c) s += wr[3 + c] * pos[((long)b * 3 + c) * NN + n];
  out[i] = s;
}

__global__ void fc_out_k(const float* __restrict__ x4, const float* __restrict__ w,
                         const float* __restrict__ bias, float* __restrict__ out)
{
  const int idx = blockIdx.x * 256 + threadIdx.x;            // over B*NN
  const int n = idx % NN;
  const int b = idx / NN;
  float s = bias[0];
  for (int c = 0; c < CC; ++c) s += w[c] * x4[((long)b * CC + c) * NN + n];
  out[idx] = s;
}

__global__ void ewise_k(float* __restrict__ out, const float* __restrict__ a,
                        const float* __restrict__ b, int mode)                 // 0:add 1:sub
{
  const long i = (long)blockIdx.x * 256 + threadIdx.x;
  out[i] = (mode == 0) ? (a[i] + b[i]) : (a[i] - b[i]);
}

__global__ __launch_bounds__(256) void softmax_rows_k(float* __restrict__ S)
{
  const int t = threadIdx.x;
  float* p = S + (long)blockIdx.x * NN;
  __shared__ float red[256];
  float v[NN / 256];
  float mx = -3.4e38f;
#pragma unroll
  for (int i = 0; i < NN / 256; ++i) { v[i] = p[t + i * 256]; mx = fmaxf(mx, v[i]); }
  red[t] = mx; __syncthreads();
  for (int s = 128; s > 0; s >>= 1) { if (t < s) red[t] = fmaxf(red[t], red[t + s]); __syncthreads(); }
  mx = red[0]; __syncthreads();
  float sum = 0.f;
#pragma unroll
  for (int i = 0; i < NN / 256; ++i) { v[i] = expf(v[i] - mx); sum += v[i]; }
  red[t] = sum; __syncthreads();
  for (int s = 128; s > 0; s >>= 1) { if (t < s) red[t] += red[t + s]; __syncthreads(); }
  const float inv = 1.f / red[0];
#pragma unroll
  for (int i = 0; i < NN / 256; ++i) p[t + i * 256] = v[i] * inv;
}

__global__ void colsum_scale_k(const float* __restrict__ S, float* __restrict__ scale)
{
  const int m = blockIdx.x * 256 + threadIdx.x;
  float s = 0.f;
  for (int n = 0; n < NN; ++n) s += S[(long)n * NN + m];
  scale[m] = 1.f / (1e-9f + s);
}

__global__ __launch_bounds__(256) void bn_stats_k(const float* __restrict__ X,
                                                  float* __restrict__ mean,
                                                  float* __restrict__ istd)
{
  const int c = blockIdx.x, t = threadIdx.x;
  float s = 0.f, s2 = 0.f;
  for (int b = 0; b < BB; ++b) {
    const float* p = X + ((long)b * CC + c) * NN;
    for (int n = t; n < NN; n += 256) { float v = p[n]; s += v; s2 += v * v; }
  }
  __shared__ float r1[256], r2[256];
  r1[t] = s; r2[t] = s2; __syncthreads();
  for (int k = 128; k > 0; k >>= 1) { if (t < k) { r1[t] += r1[t + k]; r2[t] += r2[t + k]; } __syncthreads(); }
  if (t == 0) {
    const float inv = 1.f / (float)(BB * NN);
    const float m = r1[0] * inv;
    const float var = r2[0] * inv - m * m;
    mean[c] = m; istd[c] = rsqrtf(var + 1e-5f);
  }
}

__global__ void bn_apply_k(float* __restrict__ out, const float* __restrict__ x,
                           const float* __restrict__ mean, const float* __restrict__ istd,
                           const float* __restrict__ g, const float* __restrict__ b,
                           const float* __restrict__ res, int mode)  // 0:plain 1:relu 2:res+relu
{
  const long i = (long)blockIdx.x * 256 + threadIdx.x;
  const int c = (int)((i / NN) % CC);
  float v = (x[i] - mean[c]) * istd[c] * g[c] + b[c];
  if (mode >= 1) v = fmaxf(v, 0.f);
  if (mode == 2) v += res[i];
  out[i] = v;
}

__global__ void gca_combine_k(float* __restrict__ out, const float* __restrict__ rr,
                              const float* __restrict__ mean, const float* __restrict__ istd,
                              const float* __restrict__ gr, const float* __restrict__ br,
                              const float* __restrict__ lmbn, const float* __restrict__ gp)
{
  const long i = (long)blockIdx.x * 256 + threadIdx.x;
  const int c = (int)((i / NN) % CC);
  const int b = (int)(i / ((long)CC * NN));
  const float v = (rr[i] - mean[c]) * istd[c] * gr[c] + br[c] + lmbn[i] * gp[b * CC + c];
  out[i] = fmaxf(v, 0.f);
}

__global__ __launch_bounds__(256) void mean_pool_k(const float* __restrict__ X,
                                                   float* __restrict__ gp0)
{
  const int c = blockIdx.x, b = blockIdx.y, t = threadIdx.x;
  const float* p = X + ((long)b * CC + c) * NN;
  float s = 0.f;
  for (int n = t; n < NN; n += 256) s += p[n];
  __shared__ float red[256];
  red[t] = s; __syncthreads();
  for (int k = 128; k > 0; k >>= 1) { if (t < k) red[t] += red[t + k]; __syncthreads(); }
  if (t == 0) gp0[b * CC + c] = red[0] * (1.f / (float)NN);
}

__global__ void gca_fc_k(const float* __restrict__ w1, const float* __restrict__ gp0,
                         float* __restrict__ gp1)
{
  const int idx = blockIdx.x * 256 + threadIdx.x;  // over BB*CC
  const int c = idx % CC, b = idx / CC;
  float s = 0.f;
  for (int k = 0; k < CC; ++k) s += w1[c * CC + k] * gp0[b * CC + k];
  gp1[idx] = s;
}

__global__ void gca_bn_small_k(const float* __restrict__ gp1, float* __restrict__ gp,
                               const float* __restrict__ gh, const float* __restrict__ bh)
{
  const int c = threadIdx.x;  // 256 threads
  float m = 0.f;
  for (int b = 0; b < BB; ++b) m += gp1[b * CC + c];
  m *= (1.f / BB);
  float v = 0.f;
  for (int b = 0; b < BB; ++b) { const float d = gp1[b * CC + c] - m; v += d * d; }
  v *= (1.f / BB);
  const float is = rsqrtf(v + 1e-5f);
  for (int b = 0; b < BB; ++b)
    gp[b * CC + c] = fmaxf(0.f, (gp1[b * CC + c] - m) * is * gh[c] + bh[c]);
}

__global__ __launch_bounds__(256) void maxpool_bcast_k(float* __restrict__ H)
{
  const int c = blockIdx.x, b = blockIdx.y, t = threadIdx.x;   // c in [0,HH)
  float* src = H + ((long)b * CC + c) * NN;
  float mx = -3.4e38f;
  for (int n = t; n < NN; n += 256) mx = fmaxf(mx, src[n]);
  __shared__ float red[256];
  red[t] = mx; __syncthreads();
  for (int k = 128; k > 0; k >>= 1) { if (t < k) red[t] = fmaxf(red[t], red[t + k]); __syncthreads(); }
  mx = red[0];
  float* dst = src + (long)HH * NN;
  for (int n = t; n < NN; n += 256) dst[n] = mx;
}

// ---------------------------------------------------------------------------
// Host orchestration
// ---------------------------------------------------------------------------
extern "C" void kernel_launch(void* const* d_in, const int* in_sizes, int n_in,
                              void* d_out, int out_size, void* d_ws, size_t ws_size,
                              hipStream_t stream)
{
  (void)in_sizes; (void)n_in; (void)out_size; (void)ws_size;

  const float* xin = (const float*)d_in[0];
  const float* pos = (const float*)d_in[1];
  auto P = [&](int i) { return (const float*)d_in[i]; };

  const long F    = (long)BB * CC * NN;   // 4,194,304 floats
  const long HALF = F / 2;
  const long ESZ  = (long)BB * CQ * NN;

  float* W  = (float*)d_ws;               // workspace needs ~206 MB
  float* Hb = W + 0 * F;
  float* X0 = W + 1 * F;
  float* X1 = W + 2 * F;
  float* X2 = W + 3 * F;
  float* X3 = W + 4 * F;
  float* GC = W + 5 * F;
  float* FU = W + 6 * F;
  float* Vb = W + 7 * F;                  // aliases GCA G1
  float* XR = W + 8 * F;                  // aliases GCA G2
  float* Ub = W + 9 * F;
  float* Sb = W + 10 * F;                 // per-batch attention matrix [NN,NN]
  float* Eb = W + 11 * F;                 // [BB,CQ,NN]
  float* T1 = Eb + ESZ;                   // [BB,HH,NN]
  float* SH = T1 + HALF;                  // [BB,HH,NN]
  float* SM = SH + HALF;                  // small scratch
  float* MEAN = SM;
  float* ISTD = SM + 256;
  float* SCL  = SM + 512;                 // 2048
  float* GP0  = SM + 512 + 2048;
  float* GP1  = GP0 + BB * CC;
  float* GP   = GP1 + BB * CC;

  auto G = [&](const float* A, const float* Bm, float* Y, const float* bias,
               const float* addt, const float* scale, int M, int K, int lda, int atr,
               long a_bs, long b_bs, long y_bs, long add_bs, int epi, int nb) {
    wgemm_f32<<<dim3(NN / 64, M / 64, nb), 128, 0, stream>>>(
        A, Bm, Y, bias, addt, scale, K, lda, atr, a_bs, b_bs, y_bs, add_bs, epi);
  };

  const int NB = (int)(F / 256);

  auto run_block = [&](const float* in, float* dest, int bi) {
    const int bse = 4 + bi * 13;
    const float *w0 = P(bse), *b0 = P(bse + 1), *w1 = P(bse + 2), *b1 = P(bse + 3),
                *ws = P(bse + 4), *bs = P(bse + 5);
    G(ws, in, SH, bs, nullptr, nullptr, HH, CC, CC, 0, 0, (long)CC * NN, (long)HH * NN, 0, EPI_BIAS, BB);
    G(w0, in, T1, b0, nullptr, nullptr, HH, CC, CC, 0, 0, (long)CC * NN, (long)HH * NN, 0, EPI_BIAS_RELU, BB);
    G(w1, T1, dest, b1, SH, nullptr, HH, HH, HH, 0, 0, (long)HH * NN, (long)CC * NN, (long)HH * NN, EPI_BIAS_ADD_RELU, BB);
    maxpool_bcast_k<<<dim3(HH, BB), 256, 0, stream>>>(dest);
  };

  auto run_sa = [&](const float* in, float* out, int si) {
    const int bse = 4 + si * 13 + 6;
    const float *wqk = P(bse), *wv = P(bse + 1), *bv = P(bse + 2),
                *wt = P(bse + 3), *bt = P(bse + 4), *gg = P(bse + 5), *bet = P(bse + 6);
    G(wqk, in, Eb, nullptr, nullptr, nullptr, CQ, CC, CC, 0, 0, (long)CC * NN, (long)CQ * NN, 0, EPI_NONE, BB);
    G(wv, in, Vb, bv, nullptr, nullptr, CC, CC, CC, 0, 0, (long)CC * NN, (long)CC * NN, 0, EPI_BIAS, BB);
    for (int b = 0; b < BB; ++b) {
      const float* E_b = Eb + (long)b * CQ * NN;
      const float* V_b = Vb + (long)b * CC * NN;
      float* XR_b = XR + (long)b * CC * NN;
      // S = E^T E  (A transposed view of E, lda = NN)
      G(E_b, E_b, Sb, nullptr, nullptr, nullptr, NN, CQ, NN, 1, 0, 0, 0, 0, EPI_NONE, 1);
      softmax_rows_k<<<NN, 256, 0, stream>>>(Sb);
      colsum_scale_k<<<NN / 256, 256, 0, stream>>>(Sb, SCL);
      // x_r = (V * att) / (1e-9 + colsum)  — column scale fused in epilogue
      G(V_b, Sb, XR_b, nullptr, nullptr, SCL, CC, NN, NN, 0, 0, 0, 0, 0, EPI_SCALE_COL, 1);
    }
    ewise_k<<<NB, 256, 0, stream>>>(XR, in, XR, 1);            // XR = in - x_r
    G(wt, XR, Ub, bt, nullptr, nullptr, CC, CC, CC, 0, 0, (long)CC * NN, (long)CC * NN, 0, EPI_BIAS, BB);
    bn_stats_k<<<CC, 256, 0, stream>>>(Ub, MEAN, ISTD);
    bn_apply_k<<<NB, 256, 0, stream>>>(out, Ub, MEAN, ISTD, gg, bet, in, 2);  // out = in + relu(bn)
  };

  auto run_gca = [&](const float* high, const float* low, float* out, int gi) {
    const int bse = 69 + (gi - 1) * 9;
    const float *w3 = P(bse), *gl = P(bse + 1), *bl = P(bse + 2),
                *w1g = P(bse + 3), *gh = P(bse + 4), *bh = P(bse + 5),
                *wr = P(bse + 6), *gr = P(bse + 7), *br = P(bse + 8);
    float* G1 = Vb;   // alias (SA temporaries dead here)
    float* G2 = XR;
    mean_pool_k<<<dim3(CC, BB), 256, 0, stream>>>(high, GP0);
    gca_fc_k<<<(BB * CC) / 256, 256, 0, stream>>>(w1g, GP0, GP1);
    gca_bn_small_k<<<1, CC, 0, stream>>>(GP1, GP, gh, bh);
    wgemm_conv3<<<dim3(NN / 64, CC / 64, BB), 128, 0, stream>>>(w3, low, G1);
    bn_stats_k<<<CC, 256, 0, stream>>>(G1, MEAN, ISTD);
    bn_apply_k<<<NB, 256, 0, stream>>>(G1, G1, MEAN, ISTD, gl, bl, nullptr, 0);
    G(wr, high, G2, nullptr, nullptr, nullptr, CC, CC, CC, 0, 0, (long)CC * NN, (long)CC * NN, 0, EPI_NONE, BB);
    bn_stats_k<<<CC, 256, 0, stream>>>(G2, MEAN, ISTD);
    gca_combine_k<<<NB, 256, 0, stream>>>(out, G2, MEAN, ISTD, gr, br, G1, GP);
  };

  // ------------------------------- forward --------------------------------
  fc_in_k<<<NB, 256, 0, stream>>>(xin, pos, P(2), P(3), FU);   // FU = fc_in(cat(x,pos))
  run_block(FU, Hb, 0);                                        // h  = catpool(block0)
  run_sa(Hb, X0, 0);                                           // x0
  run_block(X0, Hb, 1);
  run_sa(Hb, X1, 1);                                           // x1
  run_gca(X1, X0, GC, 1);                                      // x01
  run_block(X1, Hb, 2);
  ewise_k<<<NB, 256, 0, stream>>>(FU, Hb, GC, 0);              // FU = h + x01
  run_sa(FU, X2, 2);                                           // x2
  run_gca(X2, X1, GC, 2);                                      // x12
  run_block(X2, Hb, 3);
  ewise_k<<<NB, 256, 0, stream>>>(FU, Hb, GC, 0);              // FU = h + x12
  run_sa(FU, X3, 3);                                           // x3
  run_gca(X3, X2, GC, 3);                                      // x23
  run_block(Hb, X0, 4);                                        // h4 (from pre-fusion h!) -> X0 buf
  ewise_k<<<NB, 256, 0, stream>>>(FU, X0, GC, 0);              // FU = h4 + x23
  run_sa(FU, X1, 4);                                           // x4 -> X1 buf
  fc_out_k<<<(BB * NN) / 256, 256, 0, stream>>>(X1, P(96), P(97), (float*)d_out);
}